// ColorRestoration_38139309589170
// MI455X (gfx1250) — compile-verified
//
#include <hip/hip_runtime.h>
#include <stdint.h>

// Problem constants (fixed by setup_inputs in the reference).
#define WIDTH   3072
#define HEIGHT  2048
#define WC      14          // window size
#define NT      256         // threads per block
#define PER     4           // outputs per thread
#define TILE    (NT * PER)  // 1024 columns per block
#define HALO    16          // left halo in LDS (>= 13, 16B aligned)
#define SXN     (TILE + 2 * HALO)  // 1056 floats of x staged (264 float4)
#define SZN     (TILE + HALO)      // 1040 floats of z staged (260 float4)

// Native 4-float vector (clang ext_vector) — required by the nontemporal
// builtins; 16-byte aligned, lowers to b128 stores.
typedef __attribute__((ext_vector_type(4))) float v4f;

// CDNA5 async global->LDS DMA: one 16B chunk per active lane.
// %0 = LDS byte offset (low 32 bits of the generic shared pointer),
// %1 = 64-bit global address, GV addressing (saddr = off).
__device__ __forceinline__ void async_ld_b128(void* ldsp, const void* gp) {
  asm volatile("global_load_async_to_lds_b128 %0, %1, off"
               :: "v"((uint32_t)(uintptr_t)ldsp),
                  "v"((uint64_t)(uintptr_t)gp)
               : "memory");
}

__device__ __forceinline__ void wait_async0() {
#if __has_builtin(__builtin_amdgcn_s_wait_asynccnt)
  __builtin_amdgcn_s_wait_asynccnt(0);
#else
  asm volatile("s_wait_asynccnt 0" ::: "memory");
#endif
}

// Stage one 4-float chunk at local index i (chunk-aligned): async DMA when the
// whole chunk is in-bounds, scalar zero-extended fallback at the image edges.
__device__ __forceinline__ void stage_chunk(float* lds, const float* rowp,
                                            int i, int w0) {
  const int g = w0 - HALO + i;
  if (g >= 0 && g + 3 < WIDTH) {
    async_ld_b128(&lds[i], rowp + g);
  } else {
#pragma unroll
    for (int u = 0; u < 4; ++u) {
      const int gg = g + u;
      lds[i + u] = (gg >= 0 && gg < WIDTH) ? rowp[gg] : 0.0f;  // zero-extend
    }
  }
}

__global__ __launch_bounds__(NT)
void ColorRestoration_kernel(const float* __restrict__ x,
                             const float* __restrict__ z,
                             float* __restrict__ out) {
  __shared__ float sx[SXN];
  __shared__ float sz[SZN];

  const int t  = threadIdx.x;
  const int h  = blockIdx.y;
  const int w0 = blockIdx.x * TILE;
  const float* xr = x + (size_t)h * WIDTH;
  const float* zr = z + (size_t)h * WIDTH;

  // ---- Stage tiles (with halo) into LDS via async DMA --------------------
  // SXN/4 = 264 = 256 + 8 chunks; SZN/4 = 260 = 256 + 4 chunks.
  stage_chunk(sx, xr, t * 4, w0);                       // full round
  if (t < (SXN / 4 - NT)) stage_chunk(sx, xr, (NT + t) * 4, w0);  // tail
  stage_chunk(sz, zr, t * 4, w0);                       // full round
  if (t < (SZN / 4 - NT)) stage_chunk(sz, zr, (NT + t) * 4, w0);  // tail

  wait_async0();     // this wave's async transfers landed in LDS
  __syncthreads();   // all waves' transfers visible

  // ---- Compute: sliding 14-tap window, 4 consecutive columns/thread ------
  // Local index of first output column w = w0 + t*PER is li0 (sx[li0] == x[w]).
  const int li0 = t * PER + HALO;

  float den = 0.0f, n0 = 0.0f, n1 = 0.0f, n2 = 0.0f;
#pragma unroll
  for (int d = 0; d < WC; ++d) {
    const int j = li0 - d;         // window element j = w - d
    const float zv = sz[j];
    den += zv;
    n0 = fmaf(sx[j + 3],  zv, n0);
    n1 = fmaf(sx[j + 7],  zv, n1);
    n2 = fmaf(sx[j + 10], zv, n2);
  }

  float yacc[3][PER];
  float racc[3][PER];
#pragma unroll
  for (int k = 0; k < PER; ++k) {
    const int li = li0 + k;
    if (k > 0) {
      // Slide window: add product at w, drop product at w-14.
      const float zin = sz[li];
      const float zot = sz[li - WC];
      den += zin - zot;
      n0 += sx[li + 3]  * zin - sx[li - WC + 3]  * zot;
      n1 += sx[li + 7]  * zin - sx[li - WC + 7]  * zot;
      n2 += sx[li + 10] * zin - sx[li - WC + 10] * zot;
    }
    const float inv = 1.0f / den;   // denominator shared across channels
    yacc[0][k] = n0 * inv;
    yacc[1][k] = n1 * inv;
    yacc[2][k] = n2 * inv;
    racc[0][k] = sz[li - 3];        // z[w - 3]
    racc[1][k] = sz[li - 7];        // z[w - 7]
    racc[2][k] = sz[li - 10];       // z[w - 10]
  }

  // ---- Coalesced non-temporal b128 stores ---------------------------------
  // Outputs (151 MB) are written once, never re-read: stream them (TH=NT) so
  // x/z (50 MB) stay resident in the 192 MB L2 for halo/row reuse.
  const size_t CH   = (size_t)HEIGHT * WIDTH;
  const size_t base = (size_t)h * WIDTH + (size_t)(w0 + t * PER);
  float* __restrict__ yout = out;            // y tensor
  float* __restrict__ rout = out + 3 * CH;   // rgb_filter_image tensor
#pragma unroll
  for (int c = 0; c < 3; ++c) {
    v4f yv = {yacc[c][0], yacc[c][1], yacc[c][2], yacc[c][3]};
    v4f rv = {racc[c][0], racc[c][1], racc[c][2], racc[c][3]};
    __builtin_nontemporal_store(yv, (v4f*)(yout + c * CH + base));
    __builtin_nontemporal_store(rv, (v4f*)(rout + c * CH + base));
  }
}

extern "C" void kernel_launch(void* const* d_in, const int* in_sizes, int n_in,
                              void* d_out, int out_size, void* d_ws, size_t ws_size,
                              hipStream_t stream) {
  const float* x = (const float*)d_in[0];   // (1,1,2048,3072) f32
  const float* z = (const float*)d_in[1];   // (1,1,2048,3072) f32
  float* out = (float*)d_out;               // y (3*H*W) ++ rgb_filter (3*H*W)

  dim3 grid(WIDTH / TILE, HEIGHT);          // (3, 2048)
  dim3 block(NT);
  ColorRestoration_kernel<<<grid, block, 0, stream>>>(x, z, out);
}